// ChirpletSynth_5050881540236
// MI455X (gfx1250) — compile-verified
//
#include <hip/hip_runtime.h>

// ---------------------------------------------------------------------------
// Chirplet synth, MI455X (gfx1250, wave32).
//
// out[b, n] = sin(2*pi*phi) * fm * exp(-w^2) * sin(pi*am*t)
//   t   = (n - 32768)/44100
//   phi = (440/(fm*ln2)) * (exp2(fm*t) - 1)
//   w   = (n - 32767.5) * fm / (0.1*44100*sqrt(2))
//   am  = exp2(2*theta_am + 2)   in [4,16] Hz
//   fm  = exp2(3*theta_fm - 1)   in [0.5,4] oct/s
//
// Roofline: 64 MiB streamed output -> 2.9 us floor @ 23.3 TB/s; ~54M TRANS
// ops (v_sin/v_exp) dominate -> TRANS-bound, WMMA inapplicable (elementwise,
// no reduction). Per-row params are wave-uniform -> scalar loads +
// v_s_exp_f32 scalar transcendentals. 4 samples/thread, b128 stores.
// ---------------------------------------------------------------------------

#define N_SAMPLES          65536
#define THREADS            256
#define SAMPLES_PER_THREAD 4
#define SAMPLES_PER_BLOCK  (THREADS * SAMPLES_PER_THREAD)  // 1024

__global__ __launch_bounds__(THREADS)
void chirplet_synth_kernel(const float* __restrict__ theta_am,
                           const float* __restrict__ theta_fm,
                           float* __restrict__ out)
{
    const int b = blockIdx.y;

    // ---- per-row uniform parameters (wave-uniform index -> scalar loads) ----
    const float th_am = theta_am[b];
    const float th_fm = theta_fm[b];

    const float am = __builtin_amdgcn_exp2f(th_am * 2.0f + 2.0f);  // [4,16]
    const float fm = __builtin_amdgcn_exp2f(th_fm * 3.0f - 1.0f);  // [0.5,4]

    const float inv_sr    = 1.0f / 44100.0f;
    // F0/(fm*ln2) via hardware reciprocal (1 ulp): avoids the full IEEE
    // v_div_scale/v_div_fmas sequence; phi only feeds sin after mod-1.
    const float phi_scale = 634.7321698f * __builtin_amdgcn_rcpf(fm);
    const float w_coef    = fm * 1.60341678e-4f;        // fm / (0.1*44100*sqrt(2))
    const float mod_w     = 3.14159265358979f * am;     // pi*am rad/s (|arg| < 38: ok for v_sin)
    const float step      = __builtin_amdgcn_exp2f(fm * inv_sr);  // exp2(fm * dt)

    // ---- this thread's 4 consecutive samples ----
    const int n0 = (blockIdx.x * THREADS + threadIdx.x) * SAMPLES_PER_THREAD;
    const float t0 = ((float)n0 - 32768.0f) * inv_sr;

    // Chained chirp exponential: one v_exp_f32 + 3 multiplies instead of 4 exps.
    float e = __builtin_amdgcn_exp2f(fm * t0);

    float acc[SAMPLES_PER_THREAD];
#pragma unroll
    for (int i = 0; i < SAMPLES_PER_THREAD; ++i) {
        const float fn = (float)(n0 + i);
        const float t  = (fn - 32768.0f) * inv_sr;

        // carrier: sin(2*pi*phi); phi has period 1 in revolutions, reduce first
        // (phi can reach ~1100 rev; v_sin_f32 hw reduction only valid to 256*pi rad)
        const float phi = phi_scale * (e - 1.0f);
        e *= step;
        const float pr = phi - __builtin_rintf(phi);        // v_rndne_f32, pr in [-0.5, 0.5]
        const float carrier = __builtin_amdgcn_sinf(6.28318530717959f * pr);

        // modulator: sin(pi*am*t), argument well inside v_sin_f32's valid range
        const float modulator = __builtin_amdgcn_sinf(mod_w * t);

        // Gaussian window: exp(-w^2) = exp2(-log2(e) * w^2); underflow->0 matches ref
        const float w = (fn - 32767.5f) * w_coef;
        const float window = __builtin_amdgcn_exp2f(-1.44269504088896f * (w * w));

        acc[i] = carrier * fm * window * modulator;
    }

    // Coalesced 128-bit store (global_store_b128); n0 is 16B-aligned.
    float4 r;
    r.x = acc[0]; r.y = acc[1]; r.z = acc[2]; r.w = acc[3];
    *reinterpret_cast<float4*>(out + (size_t)b * N_SAMPLES + n0) = r;
}

extern "C" void kernel_launch(void* const* d_in, const int* in_sizes, int n_in,
                              void* d_out, int out_size, void* d_ws, size_t ws_size,
                              hipStream_t stream)
{
    const float* theta_am = (const float*)d_in[0];  // (B,) float32
    const float* theta_fm = (const float*)d_in[1];  // (B,) float32
    // d_in[2] = seed (unused by the reference synth)
    float* out = (float*)d_out;                     // (B, 1, 65536) float32

    const int B = in_sizes[0];
    dim3 grid(N_SAMPLES / SAMPLES_PER_BLOCK, B, 1); // (64, B)
    dim3 block(THREADS, 1, 1);                      // 8 wave32 waves / block
    chirplet_synth_kernel<<<grid, block, 0, stream>>>(theta_am, theta_fm, out);
}